// MDNLoss_65704409694902
// MI455X (gfx1250) — compile-verified
//
#include <hip/hip_runtime.h>
#include <cmath>

#define N_MELC 80
#define Bn 16
#define Ln 160
#define Tn 800
#define NEGV (-1e30f)
// C0 = -0.5 * N_MEL * log(2*pi)
#define C0 (-73.51508265637383f)

typedef __attribute__((ext_vector_type(2))) float v2f;
typedef __attribute__((ext_vector_type(8))) float v8f;

__device__ __forceinline__ v8f wmma_f32_16x16x4(v2f a, v2f b, v8f c) {
#if defined(__gfx1250__) && __has_builtin(__builtin_amdgcn_wmma_f32_16x16x4_f32)
  // 8 args: (neg_a, A, neg_b, B, c_mod, C, reuse_a, reuse_b)
  return __builtin_amdgcn_wmma_f32_16x16x4_f32(false, a, false, b, (short)0, c,
                                               false, false);
#else
  // host-pass / fallback stub (never executed on gfx1250 device)
  c[0] += a[0] * b[0] + a[1] * b[1];
  return c;
#endif
}

__device__ __forceinline__ float sigmoidf_(float x) {
  return 1.0f / (1.0f + __expf(-x));
}

// -----------------------------------------------------------------------------
// Kernel 1: log_prob_matrix via fp32 WMMA GEMM.
//   D[l,t] = sum_k W[l,k] * X[k,t],  K = 160 = [x ; x^2] halves
//   lp[b,l,t] = D[l,t] + c[l]   written to out[1 + b*L*T + l*T + t]
// Block: 256 threads (8 waves), one block per (batch, 16-wide l strip).
// -----------------------------------------------------------------------------
__global__ void __launch_bounds__(256) mdn_logprob_wmma(
    const float* __restrict__ mu_sigma,  // (B, L, 2*80)
    const float* __restrict__ melspec,   // (B, 80, T)
    float* __restrict__ out)             // [0]=loss, [1..]=log_prob (B,L,T)
{
  __shared__ float Wl[2 * N_MELC][16];  // W[k][l-l0]
  __shared__ float cl[16];

  const int tid = threadIdx.x;
  const int b   = blockIdx.x / (Ln / 16);
  const int l0  = (blockIdx.x % (Ln / 16)) * 16;
  const float* msb = mu_sigma + (size_t)b * Ln * 2 * N_MELC;

  // Build W tile: k<80 -> mu*iv ; k>=80 -> -0.5*iv
  for (int idx = tid; idx < 2 * N_MELC * 16; idx += 256) {
    const int k = idx >> 4, col = idx & 15;
    const float* row = msb + (l0 + col) * (2 * N_MELC);
    float w;
    if (k < N_MELC) {
      const float mu = sigmoidf_(row[k]);
      const float iv = __expf(-2.0f * row[N_MELC + k]);
      w = mu * iv;
    } else {
      const float iv = __expf(-2.0f * row[N_MELC + (k - N_MELC)]);
      w = -0.5f * iv;
    }
    Wl[k][col] = w;
  }
  // Per-l constant c[l]
  if (tid < 16) {
    const float* row = msb + (l0 + tid) * (2 * N_MELC);
    float acc = 0.f;
    for (int m = 0; m < N_MELC; ++m) {
      const float mu = sigmoidf_(row[m]);
      const float ls = row[N_MELC + m];
      const float iv = __expf(-2.0f * ls);
      acc += mu * mu * iv + ls;
    }
    cl[tid] = C0 - 0.5f * acc;
  }
  __syncthreads();

  const int wave = tid >> 5, lane = tid & 31;
  const int li = lane & 15, half = lane >> 4;  // A/B lane split per ISA layout
  const float* xb = melspec + (size_t)b * N_MELC * Tn;
  float* outp = out + 1 + (size_t)b * Ln * Tn;

  for (int t0 = wave * 16; t0 < Tn; t0 += 8 * 16) {
    const int t = t0 + li;  // N-dim coordinate (contiguous in memory)
    v8f acc = {};
    // K-half 1: X = x
    #pragma unroll
    for (int kk = 0; kk < 20; ++kk) {
      const int kb = kk * 4 + 2 * half;  // lanes 0-15: K=kb,kb+1 ; 16-31: +2
      v2f a;  a.x = Wl[kb][li];          a.y = Wl[kb + 1][li];
      v2f bb; bb.x = xb[kb * Tn + t];    bb.y = xb[(kb + 1) * Tn + t];
      acc = wmma_f32_16x16x4(a, bb, acc);
    }
    // K-half 2: X = x^2 (wave-uniform split, EXEC stays all-ones)
    #pragma unroll
    for (int kk = 20; kk < 40; ++kk) {
      const int kb = kk * 4 + 2 * half;
      const int m  = kb - N_MELC;
      v2f a;  a.x = Wl[kb][li];  a.y = Wl[kb + 1][li];
      const float x0 = xb[m * Tn + t];
      const float x1 = xb[(m + 1) * Tn + t];
      v2f bb; bb.x = x0 * x0; bb.y = x1 * x1;
      acc = wmma_f32_16x16x4(a, bb, acc);
    }
    // C layout: vgpr v, lanes 0-15 -> M=v ; lanes 16-31 -> M=v+8
    #pragma unroll
    for (int v = 0; v < 8; ++v) {
      const int lrel = v + 8 * half;
      outp[(size_t)(l0 + lrel) * Tn + t] = acc[v] + cl[lrel];
    }
  }
}

// -----------------------------------------------------------------------------
// Kernel 2: sequential alpha recursion. One wave32 per batch; each lane owns
// 5 contiguous l values so the l-1 shift is in-register + one __shfl_up.
// -----------------------------------------------------------------------------
__device__ __forceinline__ float lae(float a, float b) {  // logaddexp
  const float m = fmaxf(a, b);
  const float d = fabsf(a - b);
  return m + log1pf(__expf(-d));
}

__global__ void __launch_bounds__(32) mdn_scan(
    const float* __restrict__ lp,        // (B, L, T)
    const int* __restrict__ text_len,
    const int* __restrict__ mel_len,
    float* __restrict__ partial)         // (B,) per-batch alpha_last
{
  const int b = blockIdx.x;
  const int lane = threadIdx.x;
  const float* lpb = lp + (size_t)b * Ln * Tn;
  const int lbase = lane * 5;

  float alpha[5];
  #pragma unroll
  for (int j = 0; j < 5; ++j) alpha[j] = NEGV;
  if (lane == 0) alpha[0] = lpb[0];  // init[b,0] = lp[b,0,0]

  const int tl = text_len[b] - 1;
  const int ml = mel_len[b] - 1;
  float last = NEGV;
  if (ml == 0 && tl >= lbase && tl < lbase + 5) last = alpha[tl - lbase];

  for (int t = 1; t < Tn; ++t) {
    float carry = __shfl_up(alpha[4], 1);  // alpha_prev[l-1] across lanes
    if (lane == 0) carry = NEGV;
    float lpv[5];
    #pragma unroll
    for (int j = 0; j < 5; ++j) lpv[j] = lpb[(size_t)(lbase + j) * Tn + t];
    const float na0 = lae(alpha[0], carry)    + lpv[0];
    const float na1 = lae(alpha[1], alpha[0]) + lpv[1];
    const float na2 = lae(alpha[2], alpha[1]) + lpv[2];
    const float na3 = lae(alpha[3], alpha[2]) + lpv[3];
    const float na4 = lae(alpha[4], alpha[3]) + lpv[4];
    alpha[0] = na0; alpha[1] = na1; alpha[2] = na2; alpha[3] = na3; alpha[4] = na4;
    if (t == ml && tl >= lbase && tl < lbase + 5) last = alpha[tl - lbase];
  }
  const float result = __shfl(last, tl / 5);
  if (lane == 0) partial[b] = result;
}

// Kernel 3: deterministic serial reduction -> loss scalar.
__global__ void mdn_final(const float* __restrict__ partial,
                          float* __restrict__ out) {
  if (threadIdx.x == 0) {
    float s = 0.f;
    for (int i = 0; i < Bn; ++i) s += partial[i];
    out[0] = -s / (float)Bn;
  }
}

extern "C" void kernel_launch(void* const* d_in, const int* in_sizes, int n_in,
                              void* d_out, int out_size, void* d_ws, size_t ws_size,
                              hipStream_t stream) {
  const float* mu_sigma = (const float*)d_in[0];   // (16,160,160)
  const float* melspec  = (const float*)d_in[1];   // (16,80,800)
  const int* text_len   = (const int*)d_in[2];     // (16,)
  const int* mel_len    = (const int*)d_in[3];     // (16,)
  float* out = (float*)d_out;                      // [0]=loss, [1..]=log_prob
  float* partial = (float*)d_ws;                   // 16 floats scratch

  mdn_logprob_wmma<<<dim3(Bn * (Ln / 16)), dim3(256), 0, stream>>>(
      mu_sigma, melspec, out);
  mdn_scan<<<dim3(Bn), dim3(32), 0, stream>>>(out + 1, text_len, mel_len,
                                              partial);
  mdn_final<<<1, 32, 0, stream>>>(partial, out);
}